// MyNLLLoss_49744311222814
// MI455X (gfx1250) — compile-verified
//
#include <hip/hip_runtime.h>
#include <stdint.h>

// Problem constants from the reference: pre[B=64, S=8192, C=512] log-softmax, y_true[B,S].
#define C_DIM 512
#define WAVES_PER_BLOCK 8
#define BLOCK_THREADS 256
#define MAX_BLOCKS 4096

// Byte offset of a __shared__ object within LDS (generic -> addrspace(3) -> int).
__device__ __forceinline__ uint32_t lds_byte_off(const void* p) {
  return (uint32_t)(uintptr_t)(__attribute__((address_space(3))) const void*)p;
}

// Stream one 2KB row (512 f32) into LDS with 4 async 128b copies.
// Per instruction: 32 lanes x 16B = 512B contiguous. ASYNCcnt += 4 (per wave).
__device__ __forceinline__ void async_row_load(uint32_t lds_base, const float* grow, int lane) {
  const float* g0 = grow + lane * 4;          // lane-contiguous 16B chunks
  uint32_t l0 = lds_base + (uint32_t)lane * 16u;
  asm volatile("global_load_async_to_lds_b128 %0, %1, off"
               :: "v"(l0),          "v"(g0)        : "memory");
  asm volatile("global_load_async_to_lds_b128 %0, %1, off"
               :: "v"(l0 + 512u),   "v"(g0 + 128)  : "memory");
  asm volatile("global_load_async_to_lds_b128 %0, %1, off"
               :: "v"(l0 + 1024u),  "v"(g0 + 256)  : "memory");
  asm volatile("global_load_async_to_lds_b128 %0, %1, off"
               :: "v"(l0 + 1536u),  "v"(g0 + 384)  : "memory");
}

__global__ __launch_bounds__(BLOCK_THREADS)
void nll_offbyone_partial(const float* __restrict__ pre, const int* __restrict__ yt,
                          float* __restrict__ partial, int nrows) {
  // Per-wave double buffer: 8 waves * 2 * 512 floats = 32 KB LDS.
  __shared__ float4 buf[WAVES_PER_BLOCK][2][C_DIM / 4];
  __shared__ float wpart[WAVES_PER_BLOCK];

  const int lane = threadIdx.x & 31;
  const int wave = threadIdx.x >> 5;
  const int gw   = blockIdx.x * WAVES_PER_BLOCK + wave;   // global wave id
  const int nw   = gridDim.x * WAVES_PER_BLOCK;           // total waves

  float acc = 0.0f;
  int row = gw;                                           // uniform within wave -> EXEC all-ones

  if (row < nrows) {
    const uint32_t lb0 = lds_byte_off(&buf[wave][0][0]);
    const uint32_t lb1 = lds_byte_off(&buf[wave][1][0]);

    // Prologue: prefetch first row into buffer 0.
    async_row_load(lb0, pre + (size_t)row * C_DIM, lane);
    int cur = 0;

    while (row < nrows) {
      const int next = row + nw;
      if (next < nrows) {
        // WAR guard: DS pipe is unordered vs ASYNC pipe; ensure prior ds_loads
        // of the buffer we are about to overwrite have fully completed.
        asm volatile("s_wait_dscnt 0x0" ::: "memory");
        async_row_load(cur ? lb0 : lb1, pre + (size_t)next * C_DIM, lane);
        asm volatile("s_wait_asynccnt 0x4" ::: "memory");  // current row's 4 copies done
      } else {
        asm volatile("s_wait_asynccnt 0x0" ::: "memory");
      }

      const float4* rv = &buf[wave][cur][0];

      // Lane-local argmax over 16 columns (4 x ds_load_b128, conflict-free).
      float m = -3.402823466e38f;
      int mi = 0;
      #pragma unroll
      for (int k = 0; k < 4; ++k) {
        const float4 v = rv[lane + 32 * k];
        const int c0 = 4 * (lane + 32 * k);
        if (v.x > m) { m = v.x; mi = c0;     }
        if (v.y > m) { m = v.y; mi = c0 + 1; }
        if (v.z > m) { m = v.z; mi = c0 + 2; }
        if (v.w > m) { m = v.w; mi = c0 + 3; }
      }

      // wave32 butterfly reduce (max, first index on ties).
      #pragma unroll
      for (int off = 16; off > 0; off >>= 1) {
        const float om = __shfl_xor(m, off, 32);
        const int   oi = __shfl_xor(mi, off, 32);
        if (om > m || (om == m && oi < mi)) { m = om; mi = oi; }
      }

      const int   y = yt[row];
      const float g = ((const float*)rv)[y];              // LDS broadcast read
      const int   d = mi - y;
      const float w = (d == 1 || d == -1) ? 2.0f : 1.0f;  // WEIGHT on off-by-one
      if (lane == 0) acc += g * w;

      row = next;
      cur ^= 1;
    }
  }

  if (lane == 0) wpart[wave] = acc;
  __syncthreads();
  if (threadIdx.x == 0) {
    float s = 0.0f;
    #pragma unroll
    for (int i = 0; i < WAVES_PER_BLOCK; ++i) s += wpart[i];
    partial[blockIdx.x] = s;
  }
}

// Single-block, fixed-order tree reduction -> deterministic across replays.
__global__ __launch_bounds__(BLOCK_THREADS)
void final_reduce(const float* __restrict__ part, int n, float* __restrict__ out, float scale) {
  __shared__ float s[BLOCK_THREADS];
  float a = 0.0f;
  for (int i = threadIdx.x; i < n; i += BLOCK_THREADS) a += part[i];
  s[threadIdx.x] = a;
  __syncthreads();
  for (int t = BLOCK_THREADS / 2; t > 0; t >>= 1) {
    if ((int)threadIdx.x < t) s[threadIdx.x] += s[threadIdx.x + t];
    __syncthreads();
  }
  if (threadIdx.x == 0) out[0] = s[0] * scale;
}

extern "C" void kernel_launch(void* const* d_in, const int* in_sizes, int n_in,
                              void* d_out, int out_size, void* d_ws, size_t ws_size,
                              hipStream_t stream) {
  const float* pre = (const float*)d_in[0];   // [B*S, 512] f32 log-softmax
  const int*   yt  = (const int*)d_in[1];     // [B*S] i32 labels
  float*       out = (float*)d_out;           // scalar f32
  float*       ws  = (float*)d_ws;            // per-block partials

  const int rows = in_sizes[1];               // B*S = 524288

  int blocks = MAX_BLOCKS;
  const size_t max_part = ws_size / sizeof(float);
  if ((size_t)blocks > max_part) blocks = (int)max_part;
  if (blocks < 1) blocks = 1;

  nll_offbyone_partial<<<blocks, BLOCK_THREADS, 0, stream>>>(pre, yt, ws, rows);
  final_reduce<<<1, BLOCK_THREADS, 0, stream>>>(ws, blocks, out, 1.0f / 64.0f);  // / B
}